// TimeAwareTransformerEncoderLayer_8091718386180
// MI455X (gfx1250) — compile-verified
//
#include <hip/hip_runtime.h>
#include <hip/hip_bf16.h>

// ---------------- shapes ----------------
#define BSZ   4
#define SEQ   512
#define DM    128
#define NHEAD 4
#define HS    32
#define DFF   512
#define NEG_INF_F (-4294967295.0f)
#define INV_SQRT_HS 0.17677669529663687f

typedef __attribute__((ext_vector_type(2))) float v2f;
typedef __attribute__((ext_vector_type(4))) float v4f;
typedef __attribute__((ext_vector_type(8))) float v8f;

// ---------------------------------------------------------------------------
// Generic WMMA f32 GEMM:  C[m][n] = sum_k A[m*lda+k] * B[n*ldb+k] (+bias)(+relu)
// One wave (32 threads) per 16x16 output tile, K-loop of v_wmma_f32_16x16x4_f32.
// A 16x4 layout: lane holds (m = lane&15, k = 2*(lane>>4) + {0,1}).
// B 4x16 layout mirrored: lane holds (n = lane&15, k = 2*(lane>>4) + {0,1}).
// ---------------------------------------------------------------------------
template<int RELU>
__global__ void wmma_gemm_f32(const float* __restrict__ A,
                              const float* __restrict__ B,
                              const float* __restrict__ bias,
                              float* __restrict__ C,
                              int M, int N, int K, int lda, int ldb, int ldc)
{
    const int tilesN = N >> 4;
    const int tile   = blockIdx.x;
    const int m0     = (tile / tilesN) << 4;
    const int n0     = (tile % tilesN) << 4;
    const int lane   = threadIdx.x;
    const int row    = lane & 15;
    const int kk     = (lane >> 4) << 1;

    v8f acc = {};
    for (int k0 = 0; k0 < K; k0 += 4) {
        v2f a, b;
        a.x = A[(size_t)(m0 + row) * lda + k0 + kk];
        a.y = A[(size_t)(m0 + row) * lda + k0 + kk + 1];
        b.x = B[(size_t)(n0 + row) * ldb + k0 + kk];
        b.y = B[(size_t)(n0 + row) * ldb + k0 + kk + 1];
        acc = __builtin_amdgcn_wmma_f32_16x16x4_f32(false, a, false, b,
                                                    (short)0, acc, false, false);
    }
    const int nOut = lane & 15;
    const int half = lane >> 4;
    #pragma unroll
    for (int v = 0; v < 8; ++v) {
        const int mOut = m0 + v + (half << 3);
        float val = acc[v];
        if (bias) val += bias[n0 + nOut];
        if (RELU) val = fmaxf(val, 0.0f);
        C[(size_t)mOut * ldc + n0 + nOut] = val;
    }
}

// ---------------------------------------------------------------------------
// Scores: S[h][q][k] = sum_{d<32} Q0[q][h*32+d] * K0[k][h*32+d]   (raw, b=0)
// grid (32, 32, NHEAD), block 32
// ---------------------------------------------------------------------------
__global__ void qk_scores_wmma(const float* __restrict__ Q0,
                               const float* __restrict__ K0,
                               float* __restrict__ S)
{
    const int h    = blockIdx.z;
    const int m0   = blockIdx.y << 4;
    const int n0   = blockIdx.x << 4;
    const int lane = threadIdx.x;
    const int row  = lane & 15;
    const int kk   = (lane >> 4) << 1;
    const float* A = Q0 + h * HS;
    const float* B = K0 + h * HS;

    v8f acc = {};
    #pragma unroll
    for (int k0 = 0; k0 < HS; k0 += 4) {
        v2f a, b;
        a.x = A[(m0 + row) * DM + k0 + kk];
        a.y = A[(m0 + row) * DM + k0 + kk + 1];
        b.x = B[(n0 + row) * DM + k0 + kk];
        b.y = B[(n0 + row) * DM + k0 + kk + 1];
        acc = __builtin_amdgcn_wmma_f32_16x16x4_f32(false, a, false, b,
                                                    (short)0, acc, false, false);
    }
    float* Sh = S + (size_t)h * SEQ * SEQ;
    const int n = lane & 15, half = lane >> 4;
    #pragma unroll
    for (int v = 0; v < 8; ++v)
        Sh[(size_t)(m0 + v + (half << 3)) * SEQ + n0 + n] = acc[v];
}

// ---------------------------------------------------------------------------
// attn @ V (b=0): O[q][h*32+n] += sum_k attn[h][q][k] * V0[k][h*32+n]
// grid (HS/16=2, SEQ/16=32, NHEAD), block 32.  B operand is K-major (V natural).
// ---------------------------------------------------------------------------
__global__ void attn_v_wmma(const float* __restrict__ S,
                            const float* __restrict__ V0,
                            float* __restrict__ O)
{
    const int h    = blockIdx.z;
    const int m0   = blockIdx.y << 4;
    const int n0   = blockIdx.x << 4;
    const int lane = threadIdx.x;
    const int row  = lane & 15;
    const int kk   = (lane >> 4) << 1;
    const float* A = S + (size_t)h * SEQ * SEQ;

    v8f acc = {};
    for (int k0 = 0; k0 < SEQ; k0 += 4) {
        v2f a, b;
        a.x = A[(size_t)(m0 + row) * SEQ + k0 + kk];
        a.y = A[(size_t)(m0 + row) * SEQ + k0 + kk + 1];
        b.x = V0[(size_t)(k0 + kk)     * DM + h * HS + n0 + row];
        b.y = V0[(size_t)(k0 + kk + 1) * DM + h * HS + n0 + row];
        acc = __builtin_amdgcn_wmma_f32_16x16x4_f32(false, a, false, b,
                                                    (short)0, acc, false, false);
    }
    const int n = lane & 15, half = lane >> 4;
    #pragma unroll
    for (int v = 0; v < 8; ++v) {
        float* p = O + (size_t)(m0 + v + (half << 3)) * DM + h * HS + n0 + n;
        *p += acc[v];
    }
}

// ---------------------------------------------------------------------------
// Fused per-(q) kernel, b=0 only. 512 blocks x 256 threads (8 waves).
// Phase 1: stream tmK[0,q,:,:] (non-temporal b128), per-head dot(q) into scores.
// Phase 2: softmax per head (one wave per head), write attn back to S.
// Phase 3: stream tmV[0,q,:,:] (non-temporal b128), O[q][:] = sum_k attn * row.
// ---------------------------------------------------------------------------
__global__ void tm_scores_softmax_tmv(const float* __restrict__ tmK,
                                      const float* __restrict__ tmV,
                                      const float* __restrict__ Q0,
                                      const unsigned char* __restrict__ kpm,
                                      float* __restrict__ S,
                                      float* __restrict__ O)
{
    __shared__ float qs[DM];
    __shared__ float attn[NHEAD][SEQ];
    __shared__ v4f   red[256];

    const int q    = blockIdx.x;
    const int tid  = threadIdx.x;
    const int lane = tid & 31;
    const int kgrp = tid >> 5;          // 0..7 : which k-row in the 8-row strip
    const int d0   = lane << 2;          // 4 floats per lane covers d0..d0+3
    const int head = lane >> 3;          // 8 lanes per head

    if (tid < DM) qs[tid] = Q0[q * DM + tid];
    __syncthreads();

    // loop-invariant q coefficients for this lane, held in registers
    const float q0 = qs[d0], q1 = qs[d0 + 1], q2 = qs[d0 + 2], q3 = qs[d0 + 3];

    const bool mq = (kpm[q] != 0);       // batch 0 mask row
    const v4f* tmk4 = (const v4f*)(tmK + (size_t)q * SEQ * DM);

    for (int it = 0; it < SEQ / 8; ++it) {
        const int k = it * 8 + kgrp;
        const v4f* rowp = tmk4 + (size_t)k * (DM / 4);
        __builtin_prefetch((const void*)(rowp + 8 * (DM / 4) + lane), 0, 0);
        const v4f v = __builtin_nontemporal_load(rowp + lane);
        float p = v.x * q0 + v.y * q1 + v.z * q2 + v.w * q3;
        p += __shfl_xor(p, 1);
        p += __shfl_xor(p, 2);
        p += __shfl_xor(p, 4);
        if ((lane & 7) == 0) {
            float s = (S[((size_t)head * SEQ + q) * SEQ + k] + p) * INV_SQRT_HS;
            if (mq && kpm[k]) s = NEG_INF_F;
            attn[head][k] = s;
        }
    }
    __syncthreads();

    // softmax: wave w handles head w (waves 4..7 idle here)
    if (tid < NHEAD * 32) {
        const int w = tid >> 5;
        float m = -3.402823466e+38f;
        #pragma unroll
        for (int j = 0; j < SEQ / 32; ++j) m = fmaxf(m, attn[w][lane + 32 * j]);
        m = fmaxf(m, __shfl_xor(m, 1));
        m = fmaxf(m, __shfl_xor(m, 2));
        m = fmaxf(m, __shfl_xor(m, 4));
        m = fmaxf(m, __shfl_xor(m, 8));
        m = fmaxf(m, __shfl_xor(m, 16));
        float sum = 0.0f;
        #pragma unroll
        for (int j = 0; j < SEQ / 32; ++j) {
            const int k = lane + 32 * j;
            const float e = __expf(attn[w][k] - m);
            attn[w][k] = e;
            sum += e;
        }
        sum += __shfl_xor(sum, 1);
        sum += __shfl_xor(sum, 2);
        sum += __shfl_xor(sum, 4);
        sum += __shfl_xor(sum, 8);
        sum += __shfl_xor(sum, 16);
        const float inv = 1.0f / sum;
        #pragma unroll
        for (int j = 0; j < SEQ / 32; ++j) {
            const int k = lane + 32 * j;
            const float a = attn[w][k] * inv;
            attn[w][k] = a;
            S[((size_t)w * SEQ + q) * SEQ + k] = a;   // attn for attn@V GEMM
        }
    }
    __syncthreads();

    // Phase 3: O[q][:] = sum_k attn[head][k] * tmV[0,q,k,:]
    const v4f* tmv4 = (const v4f*)(tmV + (size_t)q * SEQ * DM);
    v4f acc = {0.f, 0.f, 0.f, 0.f};
    for (int it = 0; it < SEQ / 8; ++it) {
        const int k = it * 8 + kgrp;
        const v4f* rowp = tmv4 + (size_t)k * (DM / 4);
        __builtin_prefetch((const void*)(rowp + 8 * (DM / 4) + lane), 0, 0);
        const v4f v = __builtin_nontemporal_load(rowp + lane);
        const float a = attn[head][k];
        acc.x += a * v.x; acc.y += a * v.y; acc.z += a * v.z; acc.w += a * v.w;
    }
    red[tid] = acc;
    __syncthreads();
    if (tid < 32) {
        v4f t = red[tid];
        #pragma unroll
        for (int j = 1; j < 8; ++j) {
            const v4f r = red[tid + 32 * j];
            t.x += r.x; t.y += r.y; t.z += r.z; t.w += r.w;
        }
        ((v4f*)(O + (size_t)q * DM))[tid] = t;
    }
}

// ---------------------------------------------------------------------------
// out[row][:] = LayerNorm( X[row][:] + R[rrow][:] ) * g + be   (d = 128)
// One wave per row; lane owns 4 contiguous elements.
// ---------------------------------------------------------------------------
__global__ void add_ln_kernel(const float* __restrict__ X,
                              const float* __restrict__ R,
                              int broadcastR,
                              const float* __restrict__ g,
                              const float* __restrict__ be,
                              float* __restrict__ out)
{
    const int row  = blockIdx.x;
    const int lane = threadIdx.x;
    const int rrow = broadcastR ? (row & (SEQ - 1)) : row;

    v4f x = ((const v4f*)(X + (size_t)row * DM))[lane];
    const v4f r = ((const v4f*)(R + (size_t)rrow * DM))[lane];
    x.x += r.x; x.y += r.y; x.z += r.z; x.w += r.w;

    float s = x.x + x.y + x.z + x.w;
    s += __shfl_xor(s, 1);  s += __shfl_xor(s, 2);  s += __shfl_xor(s, 4);
    s += __shfl_xor(s, 8);  s += __shfl_xor(s, 16);
    const float mu = s * (1.0f / DM);

    const float dx = x.x - mu, dy = x.y - mu, dz = x.z - mu, dw = x.w - mu;
    float v = dx * dx + dy * dy + dz * dz + dw * dw;
    v += __shfl_xor(v, 1);  v += __shfl_xor(v, 2);  v += __shfl_xor(v, 4);
    v += __shfl_xor(v, 8);  v += __shfl_xor(v, 16);
    const float inv = rsqrtf(v * (1.0f / DM) + 1e-5f);

    const v4f gg = ((const v4f*)g)[lane];
    const v4f bb = ((const v4f*)be)[lane];
    v4f y;
    y.x = dx * inv * gg.x + bb.x;
    y.y = dy * inv * gg.y + bb.y;
    y.z = dz * inv * gg.z + bb.z;
    y.w = dw * inv * gg.w + bb.w;
    ((v4f*)(out + (size_t)row * DM))[lane] = y;
}

// ---------------------------------------------------------------------------
extern "C" void kernel_launch(void* const* d_in, const int* in_sizes, int n_in,
                              void* d_out, int out_size, void* d_ws, size_t ws_size,
                              hipStream_t stream) {
    const float* src = (const float*)d_in[0];
    const float* tmK = (const float*)d_in[1];
    const float* tmV = (const float*)d_in[2];
    const float* Wq  = (const float*)d_in[3];
    const float* bq  = (const float*)d_in[4];
    const float* Wk  = (const float*)d_in[5];
    const float* bk  = (const float*)d_in[6];
    const float* Wv  = (const float*)d_in[7];
    const float* bv  = (const float*)d_in[8];
    const float* W1  = (const float*)d_in[9];
    const float* b1  = (const float*)d_in[10];
    const float* W2  = (const float*)d_in[11];
    const float* b2  = (const float*)d_in[12];
    const float* g1  = (const float*)d_in[13];
    const float* be1 = (const float*)d_in[14];
    const float* g2  = (const float*)d_in[15];
    const float* be2 = (const float*)d_in[16];
    const unsigned char* kpm = (const unsigned char*)d_in[17];

    // Workspace layout (floats); ~11.5 MB total.
    float* ws = (float*)d_ws;
    float* Q0 = ws;                         // [512][128]
    float* K0 = Q0 + SEQ * DM;              // [512][128]
    float* V0 = K0 + SEQ * DM;              // [512][128]
    float* S  = V0 + SEQ * DM;              // [4][512][512] scores -> attn (in place)
    float* O  = S  + (size_t)NHEAD * SEQ * SEQ; // [512][128] merged-head attn out (b=0)
    float* X1 = O  + SEQ * DM;              // [2048][128] after LN1
    float* Hf = X1 + BSZ * SEQ * DM;        // [2048][512] FFN hidden
    float* FF = Hf + (size_t)BSZ * SEQ * DFF; // [2048][128]

    const int rows = BSZ * SEQ;             // 2048

    // 1) Q/K/V projections, batch-0 rows only (first 512 rows of src).
    wmma_gemm_f32<0><<<(SEQ / 16) * (DM / 16), 32, 0, stream>>>(src, Wq, bq, Q0, SEQ, DM, DM, DM, DM, DM);
    wmma_gemm_f32<0><<<(SEQ / 16) * (DM / 16), 32, 0, stream>>>(src, Wk, bk, K0, SEQ, DM, DM, DM, DM, DM);
    wmma_gemm_f32<0><<<(SEQ / 16) * (DM / 16), 32, 0, stream>>>(src, Wv, bv, V0, SEQ, DM, DM, DM, DM, DM);

    // 2) raw Q.K^T scores per head (b=0).
    qk_scores_wmma<<<dim3(SEQ / 16, SEQ / 16, NHEAD), 32, 0, stream>>>(Q0, K0, S);

    // 3) time-matrix score term + softmax + time-matrix V term (streams 268 MB).
    tm_scores_softmax_tmv<<<SEQ, 256, 0, stream>>>(tmK, tmV, Q0, kpm, S, O);

    // 4) O += attn @ V.
    attn_v_wmma<<<dim3(HS / 16, SEQ / 16, NHEAD), 32, 0, stream>>>(S, V0, O);

    // 5) x1 = LN(src + sa) with sa = O broadcast over batch.
    add_ln_kernel<<<rows, 32, 0, stream>>>(src, O, 1, g1, be1, X1);

    // 6) FFN: hidden = relu(x1 @ W1^T + b1); ff = hidden @ W2^T + b2.
    wmma_gemm_f32<1><<<(rows / 16) * (DFF / 16), 32, 0, stream>>>(X1, W1, b1, Hf, rows, DFF, DM, DM, DM, DFF);
    wmma_gemm_f32<0><<<(rows / 16) * (DM / 16), 32, 0, stream>>>(Hf, W2, b2, FF, rows, DM, DFF, DFF, DFF, DM);

    // 7) out = LN(x1 + ff).
    add_ln_kernel<<<rows, 32, 0, stream>>>(X1, FF, 0, g2, be2, (float*)d_out);
}